// ASTGenerationHead_1958505087653
// MI455X (gfx1250) — compile-verified
//
#include <hip/hip_runtime.h>
#include <hip/hip_bf16.h>

typedef float v2f __attribute__((ext_vector_type(2)));
typedef float v8f __attribute__((ext_vector_type(8)));

#define B_   256
#define S_   128
#define H_   512
#define N_   30
#define NT_  30
#define HH_  256
#define NCOL 544               // 1 + 30 + 1 + 256 + 256 = 34 * 16

// d_out float offsets (outputs concatenated in return order)
#define OUT_NUM   0            // (256,30)
#define OUT_EXIST 7680         // (256,30)
#define OUT_TYPE  15360        // (256,30,30)
#define OUT_VAL   245760       // (256,30)
#define OUT_ADJ   253440       // (256,30,30)

// workspace float offsets
#define WS_GREPR 0             // 256*512
#define WS_AI    131072        // 7680*256
#define WS_AJ    2097152       // 7680*256

#define LDS_STRIDE 514         // 512 + 2 pad -> distinct bank pairs per A row

// ---------------------------------------------------------------------------
// Kernel 1 (fused): per-batch mean over S + head GEMM.
// grid = 256 (one block per batch), block = 256 (8 waves).
// Phase 1 streams hs[b] (128x512) once: accumulates the mean and stashes
// rows s<30 into LDS.  Phase 2 computes node_h[b] (30x512, padded to 2x16
// rows) @ Wcombined.T (512x544) with V_WMMA_F32_16X16X4_F32 out of LDS.
// columns: [0]=exist [1..30]=type [31]=val [32..287]=ai(+b_adj1) [288..543]=aj
__global__ void mean_head_kernel(const float* __restrict__ hs,
                                 const float* __restrict__ w_exist, const float* __restrict__ b_exist,
                                 const float* __restrict__ w_type,  const float* __restrict__ b_type,
                                 const float* __restrict__ w_val,   const float* __restrict__ b_val,
                                 const float* __restrict__ w_adj1,  const float* __restrict__ b_adj1,
                                 float* __restrict__ grepr,
                                 float* __restrict__ d_out,
                                 float* __restrict__ ws_ai, float* __restrict__ ws_aj) {
  __shared__ float As[31 * LDS_STRIDE];     // 62.2 KB: rows 0..29 valid, row 30 = dummy
  const int b   = blockIdx.x;
  const int tid = threadIdx.x;

  // ---- Phase 1: single pass over hs[b], mean + LDS stash -------------------
  {
    const float* p = hs + (size_t)b * S_ * H_;
    const int h0 = tid, h1 = tid + 256;
    float s0 = 0.f, s1 = 0.f;
    for (int s = 0; s < S_; ++s) {
      const float f0 = p[(size_t)s * H_ + h0];
      const float f1 = p[(size_t)s * H_ + h1];
      s0 += f0; s1 += f1;
      if (s < N_) {                         // uniform branch per iteration
        As[s * LDS_STRIDE + h0] = f0;
        As[s * LDS_STRIDE + h1] = f1;
      }
    }
    grepr[(size_t)b * H_ + h0] = s0 * (1.f / 128.f);
    grepr[(size_t)b * H_ + h1] = s1 * (1.f / 128.f);
  }
  __syncthreads();

  // ---- Phase 2: 2 Mtiles x 34 Ntiles = 68 wave jobs over 8 waves ----------
  const int wave = tid >> 5;
  const int lane = tid & 31;
  const int half = lane >> 4;               // k offset 0 / 2
  const int lo   = lane & 15;

  for (int job = wave; job < 2 * (NCOL / 16); job += 8) {
    const int mt = job & 1;
    const int nt = job >> 1;
    const int n  = nt * 16 + lo;

    const int ar = mt * 16 + lo;            // A row 0..31; clamp garbage rows
    const float* Alds = As + (size_t)((ar < 31) ? ar : 30) * LDS_STRIDE;

    const float* brow;
    if (n == 0)        brow = w_exist;
    else if (n <= 30)  brow = w_type + (size_t)(n - 1) * H_;
    else if (n == 31)  brow = w_val;
    else if (n < 288)  brow = w_adj1 + (size_t)(n - 32) * (2 * H_);
    else               brow = w_adj1 + (size_t)(n - 288) * (2 * H_) + H_;

    v8f acc = {};
    #pragma unroll 4
    for (int k0 = 0; k0 < H_; k0 += 4) {
      v2f a  = *(const v2f*)(Alds + k0 + 2 * half);
      v2f bb = *(const v2f*)(brow + k0 + 2 * half);
      acc = __builtin_amdgcn_wmma_f32_16x16x4_f32(false, a, false, bb,
                                                  (short)0, acc, false, false);
    }

    #pragma unroll
    for (int v = 0; v < 8; ++v) {
      const int i = mt * 16 + v + 8 * half; // node index within batch
      if (i < N_) {
        const int m = b * N_ + i;           // global node row
        const float x = acc[v];
        if (n == 0)       d_out[OUT_EXIST + m] = x + b_exist[0];
        else if (n <= 30) d_out[OUT_TYPE + (size_t)m * NT_ + (n - 1)] = x + b_type[n - 1];
        else if (n == 31) d_out[OUT_VAL + m] = x + b_val[0];
        else if (n < 288) ws_ai[(size_t)m * HH_ + (n - 32)]  = x + b_adj1[n - 32];
        else              ws_aj[(size_t)m * HH_ + (n - 288)] = x;
      }
    }
  }
}

// ---------------------------------------------------------------------------
// Kernel 2: num_nodes_logits = global_repr @ w_num.T + b_num  (256x512x30)
// One wave per 16x16 output tile.  grid = 16 Mtiles * 2 Ntiles, block = 32.
__global__ void numnodes_kernel(const float* __restrict__ grepr,
                                const float* __restrict__ w_num,
                                const float* __restrict__ b_num,
                                float* __restrict__ out) {
  const int mt = blockIdx.x >> 1;
  const int nt = blockIdx.x & 1;
  const int lane = threadIdx.x;
  const int half = lane >> 4;
  const int lo   = lane & 15;
  const int n = nt * 16 + lo;
  const float* arow = grepr + (size_t)(mt * 16 + lo) * H_;
  const float* brow = w_num + (size_t)((n < NT_) ? n : 0) * H_;
  v8f acc = {};
  #pragma unroll 4
  for (int k0 = 0; k0 < H_; k0 += 4) {
    v2f a  = *(const v2f*)(arow + k0 + 2 * half);
    v2f bb = *(const v2f*)(brow + k0 + 2 * half);
    acc = __builtin_amdgcn_wmma_f32_16x16x4_f32(false, a, false, bb,
                                                (short)0, acc, false, false);
  }
  if (n < NT_) {
    const float bias = b_num[n];
    #pragma unroll
    for (int v = 0; v < 8; ++v) {
      const int m = mt * 16 + v + 8 * half;
      out[OUT_NUM + (size_t)m * NT_ + n] = acc[v] + bias;
    }
  }
}

// ---------------------------------------------------------------------------
// Kernel 3: adjacency.  One block per batch, ai/aj/w2 resident in LDS.
// adj[b,i,j] = sum_k relu(ai[i,k]+aj[j,k]) * w2[k] + b2 ; diagonal = 0
__global__ void adj_kernel(const float* __restrict__ ws_ai,
                           const float* __restrict__ ws_aj,
                           const float* __restrict__ w_adj2,
                           const float* __restrict__ b_adj2,
                           float* __restrict__ d_out) {
  __shared__ float sai[N_ * HH_];
  __shared__ float saj[N_ * HH_];
  __shared__ float sw2[HH_];
  const int b   = blockIdx.x;
  const int tid = threadIdx.x;
  const float* gai = ws_ai + (size_t)b * N_ * HH_;
  const float* gaj = ws_aj + (size_t)b * N_ * HH_;
  for (int idx = tid; idx < N_ * HH_; idx += 256) {
    sai[idx] = gai[idx];
    saj[idx] = gaj[idx];
  }
  sw2[tid] = w_adj2[tid];
  __syncthreads();

  const int wave = tid >> 5;
  const int lane = tid & 31;
  const float bias = b_adj2[0];
  for (int p = wave; p < N_ * N_; p += 8) {
    const int i = p / N_;
    const int j = p - i * N_;
    const float* pa = sai + i * HH_;
    const float* pb = saj + j * HH_;
    float s = 0.f;
    #pragma unroll
    for (int t = 0; t < HH_ / 32; ++t) {
      const int k = lane + t * 32;
      float v = fmaxf(pa[k] + pb[k], 0.f);
      s = fmaf(v, sw2[k], s);
    }
    #pragma unroll
    for (int off = 16; off >= 1; off >>= 1)
      s += __shfl_xor(s, off, 32);
    if (lane == 0)
      d_out[OUT_ADJ + (size_t)b * (N_ * N_) + p] = (i == j) ? 0.f : (s + bias);
  }
}

// ---------------------------------------------------------------------------
extern "C" void kernel_launch(void* const* d_in, const int* in_sizes, int n_in,
                              void* d_out, int out_size, void* d_ws, size_t ws_size,
                              hipStream_t stream) {
  const float* hs      = (const float*)d_in[0];
  const float* w_exist = (const float*)d_in[1];
  const float* b_exist = (const float*)d_in[2];
  const float* w_type  = (const float*)d_in[3];
  const float* b_type  = (const float*)d_in[4];
  const float* w_val   = (const float*)d_in[5];
  const float* b_val   = (const float*)d_in[6];
  const float* w_adj1  = (const float*)d_in[7];
  const float* b_adj1  = (const float*)d_in[8];
  const float* w_adj2  = (const float*)d_in[9];
  const float* b_adj2  = (const float*)d_in[10];
  const float* w_num   = (const float*)d_in[11];
  const float* b_num   = (const float*)d_in[12];

  float* out   = (float*)d_out;
  float* ws    = (float*)d_ws;
  float* grepr = ws + WS_GREPR;
  float* ws_ai = ws + WS_AI;
  float* ws_aj = ws + WS_AJ;

  mean_head_kernel<<<B_, 256, 0, stream>>>(hs, w_exist, b_exist, w_type, b_type,
                                           w_val, b_val, w_adj1, b_adj1,
                                           grepr, out, ws_ai, ws_aj);
  numnodes_kernel<<<32, 32, 0, stream>>>(grepr, w_num, b_num, out);
  adj_kernel<<<B_, 256, 0, stream>>>(ws_ai, ws_aj, w_adj2, b_adj2, out);
}